// SSM_58076547776593
// MI455X (gfx1250) — compile-verified
//
#include <hip/hip_runtime.h>
#include <hip/hip_bf16.h>
#include <math.h>

#define B_SZ 2
#define LSEQ 2048
#define DDIM 512
#define NST  16

typedef __attribute__((ext_vector_type(2))) float v2f;
typedef __attribute__((ext_vector_type(8))) float v8f;

// ---------------------------------------------------------------------------
// Kernel 1: projections via V_WMMA_F32_16X16X4_F32.
// One wave (32 threads) per 16-row tile of x (rows = flattened (b,l)).
// Computes projB = x@B_w^T + B_b, projC = x@C_w^T + C_b (both (4096,16)),
// and delta = softplus(x@d_w^T + d_b)  (4096,).
// ---------------------------------------------------------------------------
__global__ __launch_bounds__(32)
void ssm_proj_wmma(const float* __restrict__ x,
                   const float* __restrict__ Bw, const float* __restrict__ Bb,
                   const float* __restrict__ Cw, const float* __restrict__ Cb,
                   const float* __restrict__ dw, const float* __restrict__ db,
                   float* __restrict__ projB, float* __restrict__ projC,
                   float* __restrict__ delta)
{
    const int lane = threadIdx.x;            // 0..31
    const int m    = lane & 15;              // row-in-tile (A) / column N (B)
    const int kh   = lane >> 4;              // which K half-pair this lane carries
    const int r0   = blockIdx.x * 16;        // first row of this tile

    v8f cB = {};   // B_in accumulator (16x16 f32, rows r0..r0+15, cols n=0..15)
    v8f cC = {};   // C_in accumulator
    float dpart = 0.0f;                      // partial of dot(x_row, d_w)

    const float* xrow = x + (size_t)(r0 + m) * DDIM;

    for (int kk = 0; kk < DDIM; kk += 4) {
        const int k0 = kk + 2 * kh;
        // A fragment: 16x4 tile of x. VGPR0 = K = 2*kh, VGPR1 = K = 2*kh+1.
        v2f av;
        av.x = xrow[k0];
        av.y = xrow[k0 + 1];

        // B fragments: 4x16 tiles, element (k,n) = W[n*D + kk + k].
        v2f bvB, bvC;
        bvB.x = Bw[(size_t)m * DDIM + k0];
        bvB.y = Bw[(size_t)m * DDIM + k0 + 1];
        bvC.x = Cw[(size_t)m * DDIM + k0];
        bvC.y = Cw[(size_t)m * DDIM + k0 + 1];

        cB = __builtin_amdgcn_wmma_f32_16x16x4_f32(
                 false, av, false, bvB, (short)0, cB, false, false);
        cC = __builtin_amdgcn_wmma_f32_16x16x4_f32(
                 false, av, false, bvC, (short)0, cC, false, false);

        // delta partial: this lane's share of dot(x_row, d_w)
        dpart = fmaf(av.x, dw[k0], dpart);
        dpart = fmaf(av.y, dw[k0 + 1], dpart);
    }

    // Store projections with bias. C/D layout: VGPR v -> row r0+v+8*kh, col m.
    const float bb = Bb[m];
    const float cb = Cb[m];
#pragma unroll
    for (int v = 0; v < 8; ++v) {
        const int row = r0 + v + 8 * kh;
        projB[(size_t)row * NST + m] = cB[v] + bb;
        projC[(size_t)row * NST + m] = cC[v] + cb;
    }

    // Reduce delta across the two K-halves (lane L <-> L+16), softplus, store.
    float dfull = dpart + __shfl_xor(dpart, 16, 32);
    if (lane < 16) {
        float raw = dfull + db[0];
        float sp  = (raw > 20.0f) ? raw : log1pf(__expf(raw));
        delta[r0 + m] = sp;
    }
}

// ---------------------------------------------------------------------------
// Kernel 2: sequential scan over L. One thread per (b,d) channel; h[16] in
// registers. projB/projC/delta are broadcast across the 512 d-threads of a
// (b,t) and stay cache-resident. -1/A hoisted out of the time loop.
// ---------------------------------------------------------------------------
__global__ __launch_bounds__(32)
void ssm_scan(const float* __restrict__ x, const float* __restrict__ A,
              const float* __restrict__ projB, const float* __restrict__ projC,
              const float* __restrict__ delta, float* __restrict__ out)
{
    const int ch = blockIdx.x * blockDim.x + threadIdx.x;   // 0..B*D-1
    if (ch >= B_SZ * DDIM) return;
    const int b = ch / DDIM;
    const int d = ch % DDIM;

    float a[NST], nrcp[NST], h[NST];
#pragma unroll
    for (int n = 0; n < NST; ++n) {
        a[n]    = A[(size_t)d * NST + n];
        nrcp[n] = (fabsf(a[n]) > 1e-6f) ? (-1.0f / a[n]) : 0.0f;
        h[n]    = 0.0f;
    }

    const float* xp = x     + (size_t)b * LSEQ * DDIM + d;
    const float* pB = projB + (size_t)b * LSEQ * NST;
    const float* pC = projC + (size_t)b * LSEQ * NST;
    const float* dl = delta + (size_t)b * LSEQ;
    float*       op = out   + (size_t)b * LSEQ * DDIM + d;

    for (int t = 0; t < LSEQ; ++t) {
        const float xt = xp[(size_t)t * DDIM];
        const float dt = dl[t];
        float y = 0.0f;
#pragma unroll
        for (int n = 0; n < NST; ++n) {
            const float an = a[n];
            const float dA = dt * an;
            const float Ad = __expf(dA);
            float frac;
            if (fabsf(an) > 1e-6f) {
                frac = (1.0f - Ad) * nrcp[n];
            } else {
                frac = dt * (1.0f - 0.5f * dA + dA * dA * (1.0f / 6.0f));
            }
            const float Bd = frac * pB[(size_t)t * NST + n];
            h[n] = fmaf(Ad, h[n], Bd * xt);
            y = fmaf(h[n], pC[(size_t)t * NST + n], y);
        }
        op[(size_t)t * DDIM] = y;
    }
}

// ---------------------------------------------------------------------------
// Launch
// ---------------------------------------------------------------------------
extern "C" void kernel_launch(void* const* d_in, const int* in_sizes, int n_in,
                              void* d_out, int out_size, void* d_ws, size_t ws_size,
                              hipStream_t stream)
{
    (void)in_sizes; (void)n_in; (void)out_size; (void)ws_size;

    const float* x   = (const float*)d_in[0];  // (B,L,D)
    const float* A   = (const float*)d_in[1];  // (D,N)
    const float* Bw  = (const float*)d_in[2];  // (N,D)
    const float* Bb  = (const float*)d_in[3];  // (N,)
    const float* Cw  = (const float*)d_in[4];  // (N,D)
    const float* Cb  = (const float*)d_in[5];  // (N,)
    const float* dw  = (const float*)d_in[6];  // (1,D)
    const float* db  = (const float*)d_in[7];  // (1,)
    float*       out = (float*)d_out;          // (B,L,D)

    const int M = B_SZ * LSEQ;                 // 4096 rows
    float* projB = (float*)d_ws;                         // M*16 floats
    float* projC = projB + (size_t)M * NST;              // M*16 floats
    float* delta = projC + (size_t)M * NST;              // M floats

    // Projections: one wave per 16-row tile.
    ssm_proj_wmma<<<M / 16, 32, 0, stream>>>(x, Bw, Bb, Cw, Cb, dw, db,
                                             projB, projC, delta);

    // Scan: 1024 channels = 32 waves, one wave per block to spread over WGPs.
    ssm_scan<<<(B_SZ * DDIM + 31) / 32, 32, 0, stream>>>(x, A, projB, projC,
                                                         delta, out);
}